// GATv2Convolution_652835029490
// MI455X (gfx1250) — compile-verified
//
#include <hip/hip_runtime.h>
#include <hip/hip_bf16.h>

typedef float vf2 __attribute__((ext_vector_type(2)));
typedef float vf8 __attribute__((ext_vector_type(8)));

#define NNODES 100000
#define NEDGES 3200000
#define ETOT   (NEDGES + NNODES)
#define FIN    165
#define HID    64
#define NCLS   2
#define NEG_SLOPE 0.2f

// ---- monotonic float <-> unsigned key for atomicMax on floats ----
static __device__ __forceinline__ unsigned fkey(float f) {
  unsigned u = __float_as_uint(f);
  return (u >> 31) ? ~u : (u ^ 0x80000000u);
}
static __device__ __forceinline__ float fdekey(unsigned k) {
  unsigned u = (k >> 31) ? (k ^ 0x80000000u) : ~k;
  return __uint_as_float(u);
}

// ---------------------------------------------------------------------------
// Layer-1 fused triple GEMM via V_WMMA_F32_16X16X4_F32:
//   XL = x@Wl1 ; XR = x@Wr1 ; H = x@Wlin1 + blin1 + b1
// One wave computes a 16-row tile across all 64 columns for all 3 weights.
// ---------------------------------------------------------------------------
__global__ __launch_bounds__(256) void gemm1_wmma(
    const float* __restrict__ x,  const float* __restrict__ Wl,
    const float* __restrict__ Wr, const float* __restrict__ Wn,
    const float* __restrict__ b1, const float* __restrict__ bn,
    float* __restrict__ XL, float* __restrict__ XR, float* __restrict__ H)
{
  const int wave = blockIdx.x * (blockDim.x >> 5) + (threadIdx.x >> 5);
  const int tiles = NNODES / 16;           // 6250, exact
  if (wave >= tiles) return;               // wave-uniform exit (EXEC all-1 inside)
  const int lane = threadIdx.x & 31;
  const int r0   = wave * 16;
  const int m    = lane & 15;              // row within A tile / col within B tile
  const int half = lane >> 4;
  const int kk   = half * 2;               // K sub-offset per ISA f32 layout
  const long rowA = (long)(r0 + m) * FIN;

  vf8 accL[4] = {}; vf8 accR[4] = {}; vf8 accN[4] = {};

  for (int ks = 0; ks < FIN - 1; ks += 4) {         // 41 full K=4 steps (k=0..163)
    vf2 a;
    a.x = x[rowA + ks + kk];
    a.y = x[rowA + ks + kk + 1];
    #pragma unroll
    for (int nt = 0; nt < 4; ++nt) {
      const int col = nt * 16 + m;
      vf2 bL, bR, bN;
      bL.x = Wl[(ks + kk) * HID + col]; bL.y = Wl[(ks + kk + 1) * HID + col];
      bR.x = Wr[(ks + kk) * HID + col]; bR.y = Wr[(ks + kk + 1) * HID + col];
      bN.x = Wn[(ks + kk) * HID + col]; bN.y = Wn[(ks + kk + 1) * HID + col];
      accL[nt] = __builtin_amdgcn_wmma_f32_16x16x4_f32(false, a, false, bL, (short)0, accL[nt], false, false);
      accR[nt] = __builtin_amdgcn_wmma_f32_16x16x4_f32(false, a, false, bR, (short)0, accR[nt], false, false);
      accN[nt] = __builtin_amdgcn_wmma_f32_16x16x4_f32(false, a, false, bN, (short)0, accN[nt], false, false);
    }
  }
  { // tail: k = 164 only (zero-pad the rest of the K=4 step)
    const int k0 = 164 + kk;
    const bool v = (k0 < FIN);
    vf2 a; a.x = v ? x[rowA + k0] : 0.f; a.y = 0.f;
    #pragma unroll
    for (int nt = 0; nt < 4; ++nt) {
      const int col = nt * 16 + m;
      vf2 bL, bR, bN;
      bL.x = v ? Wl[k0 * HID + col] : 0.f; bL.y = 0.f;
      bR.x = v ? Wr[k0 * HID + col] : 0.f; bR.y = 0.f;
      bN.x = v ? Wn[k0 * HID + col] : 0.f; bN.y = 0.f;
      accL[nt] = __builtin_amdgcn_wmma_f32_16x16x4_f32(false, a, false, bL, (short)0, accL[nt], false, false);
      accR[nt] = __builtin_amdgcn_wmma_f32_16x16x4_f32(false, a, false, bR, (short)0, accR[nt], false, false);
      accN[nt] = __builtin_amdgcn_wmma_f32_16x16x4_f32(false, a, false, bN, (short)0, accN[nt], false, false);
    }
  }
  #pragma unroll
  for (int nt = 0; nt < 4; ++nt) {
    #pragma unroll
    for (int j = 0; j < 8; ++j) {
      const int row = r0 + j + 8 * half;   // C/D layout: VGPR j -> M=j (+8 for hi lanes)
      const int col = nt * 16 + m;
      const long o = (long)row * HID + col;
      XL[o] = accL[nt][j];
      XR[o] = accR[nt][j];
      H[o]  = accN[nt][j] + b1[col] + bn[col];
    }
  }
}

// ---------------------------------------------------------------------------
__global__ void init_node(unsigned* Mx, float* Sm, unsigned* Mx2, float* Sm2) {
  int i = blockIdx.x * blockDim.x + threadIdx.x;
  if (i < NNODES) { Mx[i] = 0u; Sm[i] = 0.f; Mx2[i] = 0u; Sm2[i] = 0.f; }
}

// edge logits layer 1: one wave per edge, lanes cover dims d and d+32
__global__ __launch_bounds__(256) void edge_logits1(
    const long long* __restrict__ ei, const float* __restrict__ XL,
    const float* __restrict__ XR, const float* __restrict__ att,
    float* __restrict__ Ebuf, unsigned* __restrict__ Mx)
{
  const long ew = (long)blockIdx.x * (blockDim.x >> 5) + (threadIdx.x >> 5);
  if (ew >= ETOT) return;
  const int lane = threadIdx.x & 31;
  long s, d;
  if (ew < NEDGES) { s = (long)ei[ew]; d = (long)ei[NEDGES + ew]; }
  else             { s = d = ew - NEDGES; }
  float h0 = XL[s * HID + lane]      + XR[d * HID + lane];
  float h1 = XL[s * HID + lane + 32] + XR[d * HID + lane + 32];
  h0 = h0 > 0.f ? h0 : NEG_SLOPE * h0;
  h1 = h1 > 0.f ? h1 : NEG_SLOPE * h1;
  float p = att[lane] * h0 + att[lane + 32] * h1;
  #pragma unroll
  for (int off = 16; off > 0; off >>= 1) p += __shfl_xor(p, off, 32);
  if (lane == 0) {
    Ebuf[ew] = p;
    atomicMax(&Mx[d], fkey(p));
  }
}

// shared by both layers: w = exp(e - max[dst]); Sum[dst] += w
__global__ void edge_expsum(const long long* __restrict__ ei,
                            float* __restrict__ Ebuf,
                            const unsigned* __restrict__ Mx,
                            float* __restrict__ Sm)
{
  long i = (long)blockIdx.x * blockDim.x + threadIdx.x;
  if (i >= ETOT) return;
  long d = (i < NEDGES) ? (long)ei[NEDGES + i] : (i - NEDGES);
  float w = expf(Ebuf[i] - fdekey(Mx[d]));
  Ebuf[i] = w;
  atomicAdd(&Sm[d], w);
}

// scatter layer 1: H[dst] += alpha * XL[src], wave per edge
__global__ __launch_bounds__(256) void edge_scatter1(
    const long long* __restrict__ ei, const float* __restrict__ Ebuf,
    const float* __restrict__ Sm, const float* __restrict__ XL,
    float* __restrict__ H)
{
  const long ew = (long)blockIdx.x * (blockDim.x >> 5) + (threadIdx.x >> 5);
  if (ew >= ETOT) return;
  const int lane = threadIdx.x & 31;
  long s, d;
  if (ew < NEDGES) { s = (long)ei[ew]; d = (long)ei[NEDGES + ew]; }
  else             { s = d = ew - NEDGES; }
  float alpha = Ebuf[ew] / Sm[d];
  atomicAdd(&H[d * HID + lane],      alpha * XL[s * HID + lane]);
  atomicAdd(&H[d * HID + lane + 32], alpha * XL[s * HID + lane + 32]);
}

__global__ void elu_kernel(float* __restrict__ H) {
  long i = (long)blockIdx.x * blockDim.x + threadIdx.x;
  if (i >= (long)NNODES * HID) return;
  float v = H[i];
  H[i] = v > 0.f ? v : (expf(v) - 1.f);
}

// Layer-2 tiny GEMMs (K=64, 2 cols each) with LDS-cached weights
__global__ __launch_bounds__(256) void gemm2_kernel(
    const float* __restrict__ H, const float* __restrict__ Wl,
    const float* __restrict__ Wr, const float* __restrict__ Wn,
    const float* __restrict__ b2, const float* __restrict__ bn2,
    float* __restrict__ XL2, float* __restrict__ XR2, float* __restrict__ OUT)
{
  __shared__ float sWl[HID * 2], sWr[HID * 2], sWn[HID * 2];
  for (int t = threadIdx.x; t < HID * 2; t += blockDim.x) {
    sWl[t] = Wl[t]; sWr[t] = Wr[t]; sWn[t] = Wn[t];
  }
  __syncthreads();
  int i = blockIdx.x * blockDim.x + threadIdx.x;
  if (i >= NNODES) return;
  float l0 = 0.f, l1 = 0.f, r0 = 0.f, r1 = 0.f, o0 = 0.f, o1 = 0.f;
  const float* hr = H + (long)i * HID;
  #pragma unroll 8
  for (int k = 0; k < HID; ++k) {
    float hv = hr[k];
    l0 += hv * sWl[k * 2 + 0]; l1 += hv * sWl[k * 2 + 1];
    r0 += hv * sWr[k * 2 + 0]; r1 += hv * sWr[k * 2 + 1];
    o0 += hv * sWn[k * 2 + 0]; o1 += hv * sWn[k * 2 + 1];
  }
  XL2[i * 2] = l0; XL2[i * 2 + 1] = l1;
  XR2[i * 2] = r0; XR2[i * 2 + 1] = r1;
  OUT[i * 2]     = o0 + b2[0] + bn2[0];
  OUT[i * 2 + 1] = o1 + b2[1] + bn2[1];
}

__global__ void edge_logits2(const long long* __restrict__ ei,
                             const float* __restrict__ XL2,
                             const float* __restrict__ XR2,
                             const float* __restrict__ att,
                             float* __restrict__ Ebuf, unsigned* __restrict__ Mx)
{
  long i = (long)blockIdx.x * blockDim.x + threadIdx.x;
  if (i >= ETOT) return;
  long s, d;
  if (i < NEDGES) { s = (long)ei[i]; d = (long)ei[NEDGES + i]; }
  else            { s = d = i - NEDGES; }
  float h0 = XL2[s * 2]     + XR2[d * 2];
  float h1 = XL2[s * 2 + 1] + XR2[d * 2 + 1];
  h0 = h0 > 0.f ? h0 : NEG_SLOPE * h0;
  h1 = h1 > 0.f ? h1 : NEG_SLOPE * h1;
  float e = att[0] * h0 + att[1] * h1;
  Ebuf[i] = e;
  atomicMax(&Mx[d], fkey(e));
}

__global__ void edge_scatter2(const long long* __restrict__ ei,
                              const float* __restrict__ Ebuf,
                              const float* __restrict__ Sm,
                              const float* __restrict__ XL2,
                              float* __restrict__ OUT)
{
  long i = (long)blockIdx.x * blockDim.x + threadIdx.x;
  if (i >= ETOT) return;
  long s, d;
  if (i < NEDGES) { s = (long)ei[i]; d = (long)ei[NEDGES + i]; }
  else            { s = d = i - NEDGES; }
  float alpha = Ebuf[i] / Sm[d];
  atomicAdd(&OUT[d * 2],     alpha * XL2[s * 2]);
  atomicAdd(&OUT[d * 2 + 1], alpha * XL2[s * 2 + 1]);
}

__global__ void logsoftmax_kernel(const float* __restrict__ OUT, float* __restrict__ out) {
  int i = blockIdx.x * blockDim.x + threadIdx.x;
  if (i >= NNODES) return;
  float v0 = OUT[2 * i], v1 = OUT[2 * i + 1];
  float m = fmaxf(v0, v1);
  float l = m + logf(expf(v0 - m) + expf(v1 - m));
  out[2 * i]     = v0 - l;
  out[2 * i + 1] = v1 - l;
}

// passthrough of edge_index (second tuple output), values exact in fp32
__global__ void copy_edges(const long long* __restrict__ ei, float* __restrict__ out) {
  long i = (long)blockIdx.x * blockDim.x + threadIdx.x;
  if (i >= 2L * NEDGES) return;
  out[2L * NNODES + i] = (float)ei[i];
}

// ---------------------------------------------------------------------------
extern "C" void kernel_launch(void* const* d_in, const int* in_sizes, int n_in,
                              void* d_out, int out_size, void* d_ws, size_t ws_size,
                              hipStream_t stream) {
  const float*     x     = (const float*)d_in[0];
  const long long* ei    = (const long long*)d_in[1];
  const float*     Wl1   = (const float*)d_in[2];
  const float*     Wr1   = (const float*)d_in[3];
  const float*     att1  = (const float*)d_in[4];
  const float*     b1    = (const float*)d_in[5];
  const float*     Wlin1 = (const float*)d_in[6];
  const float*     blin1 = (const float*)d_in[7];
  const float*     Wl2   = (const float*)d_in[8];
  const float*     Wr2   = (const float*)d_in[9];
  const float*     att2  = (const float*)d_in[10];
  const float*     b2    = (const float*)d_in[11];
  const float*     Wlin2 = (const float*)d_in[12];
  const float*     blin2 = (const float*)d_in[13];
  float* out = (float*)d_out;

  // workspace layout (256B-aligned slices)
  float* ws = (float*)d_ws;
  size_t off = 0;
  auto alloc = [&](size_t nfloats) { size_t c = off; off += (nfloats + 63) & ~(size_t)63; return c; };
  float*    XL1  = ws + alloc((size_t)NNODES * HID);
  float*    XR1  = ws + alloc((size_t)NNODES * HID);
  float*    H    = ws + alloc((size_t)NNODES * HID);
  float*    Ebuf = ws + alloc((size_t)ETOT);
  unsigned* Mx1  = (unsigned*)(ws + alloc(NNODES));
  float*    Sm1  = ws + alloc(NNODES);
  float*    XL2  = ws + alloc((size_t)NNODES * 2);
  float*    XR2  = ws + alloc((size_t)NNODES * 2);
  float*    OUT2 = ws + alloc((size_t)NNODES * 2);
  unsigned* Mx2  = (unsigned*)(ws + alloc(NNODES));
  float*    Sm2  = ws + alloc(NNODES);

  const int B = 256;
  const int nodeBlocks  = (NNODES + B - 1) / B;                 // 391
  const int tileBlocks  = (NNODES / 16 + 7) / 8;                // 782 (8 waves/block)
  const int edgeWaveBlk = (ETOT + 7) / 8;                       // wave-per-edge
  const int edgeThrBlk  = (ETOT + B - 1) / B;
  const int elemBlocks  = ((long)NNODES * HID + B - 1) / B;
  const int eidxBlocks  = (int)((2L * NEDGES + B - 1) / B);

  init_node<<<nodeBlocks, B, 0, stream>>>(Mx1, Sm1, Mx2, Sm2);
  gemm1_wmma<<<tileBlocks, B, 0, stream>>>(x, Wl1, Wr1, Wlin1, b1, blin1, XL1, XR1, H);
  edge_logits1<<<edgeWaveBlk, B, 0, stream>>>(ei, XL1, XR1, att1, Ebuf, Mx1);
  edge_expsum<<<edgeThrBlk, B, 0, stream>>>(ei, Ebuf, Mx1, Sm1);
  edge_scatter1<<<edgeWaveBlk, B, 0, stream>>>(ei, Ebuf, Sm1, XL1, H);
  elu_kernel<<<elemBlocks, B, 0, stream>>>(H);
  gemm2_kernel<<<nodeBlocks, B, 0, stream>>>(H, Wl2, Wr2, Wlin2, b2, blin2, XL2, XR2, OUT2);
  edge_logits2<<<edgeThrBlk, B, 0, stream>>>(ei, XL2, XR2, att2, Ebuf, Mx2);
  edge_expsum<<<edgeThrBlk, B, 0, stream>>>(ei, Ebuf, Mx2, Sm2);
  edge_scatter2<<<edgeThrBlk, B, 0, stream>>>(ei, Ebuf, Sm2, XL2, OUT2);
  logsoftmax_kernel<<<nodeBlocks, B, 0, stream>>>(OUT2, out);
  copy_edges<<<eidxBlocks, B, 0, stream>>>(ei, out);
}